// NodeInformationScore_52312701665803
// MI455X (gfx1250) — compile-verified
//
#include <hip/hip_runtime.h>
#include <hip/hip_bf16.h>

typedef __attribute__((ext_vector_type(2))) float v2f;
typedef __attribute__((ext_vector_type(8))) float v8f;

// Kernel 1: zero the edge-accumulators and compute per-node feature row-sums
// via V_WMMA_F32_16X16X4_F32 (B = ones). One wave handles 16 node rows.
__global__ void nis_rowsum_kernel(const float* __restrict__ h,
                                  float* __restrict__ rowsum,
                                  float* __restrict__ msum,
                                  float* __restrict__ deg,
                                  int N, int D, int ntiles) {
    int gtid = blockIdx.x * blockDim.x + threadIdx.x;
    // Zero accumulators (nothing in this kernel reads them; kernel boundary
    // orders these writes before the edge kernel's atomics).
    if (gtid < N) { msum[gtid] = 0.0f; deg[gtid] = 0.0f; }

    int lane = threadIdx.x & 31;
    int tile = gtid >> 5;                 // one 16-row tile per wave
    if (tile >= ntiles) return;           // wave-uniform exit
    int base = tile * 16;

    if (base + 16 <= N) {
        // A-matrix 16x4 f32 layout (ISA 7.12.2): lanes 0-15 hold K={0,1},
        // lanes 16-31 hold K={2,3}, row M = lane&15.
        int half = lane >> 4;
        const float* hp = h + (size_t)(base + (lane & 15)) * D + 2 * half;
        v2f b; b[0] = 1.0f; b[1] = 1.0f;  // B = ones(4x16): layout-independent
        v8f acc = {0.f, 0.f, 0.f, 0.f, 0.f, 0.f, 0.f, 0.f};
        if (D == 64) {
#pragma unroll
            for (int c = 0; c < 16; ++c) {
                v2f a = *(const v2f*)(hp + 4 * c);
                acc = __builtin_amdgcn_wmma_f32_16x16x4_f32(
                    false, a, false, b, (short)0, acc, false, false);
            }
        } else {
            for (int c = 0; c < (D >> 2); ++c) {
                v2f a = *(const v2f*)(hp + 4 * c);
                acc = __builtin_amdgcn_wmma_f32_16x16x4_f32(
                    false, a, false, b, (short)0, acc, false, false);
            }
        }
        // D[m,n] is identical for all n. C/D layout: lane 0 holds M=0..7 in
        // acc[0..7]; lane 16 holds M=8..15 in acc[0..7].
        if ((lane & 15) == 0) {
            float4* o = (float4*)(rowsum + base + half * 8);
            o[0] = make_float4(acc[0], acc[1], acc[2], acc[3]);
            o[1] = make_float4(acc[4], acc[5], acc[6], acc[7]);
        }
    } else {
        // Generic tail tile (not hit for N=100000): scalar row-sums.
        int r = base + lane;
        if (lane < 16 && r < N) {
            float s = 0.0f;
            for (int c = 0; c < D; ++c) s += h[(size_t)r * D + c];
            rowsum[r] = s;
        }
    }
}

// Relaxed, agent-scope, fire-and-forget f32 atomic add -> should lower to a
// single non-returning global_atomic_add_f32 (no CAS loop).
__device__ __forceinline__ void fadd_atomic(float* p, float v) {
    (void)__hip_atomic_fetch_add(p, v, __ATOMIC_RELAXED, __HIP_MEMORY_SCOPE_AGENT);
}

// Kernel 2: per-edge gather of rowsum[src], scaled scatter-add to msum[dst],
// degree count via f32 atomics (tables are L2-resident: 3 x 400KB).
__global__ void nis_edge_kernel(const int* __restrict__ src,
                                const int* __restrict__ dst,
                                const float* __restrict__ w,
                                const float* __restrict__ rowsum,
                                float* __restrict__ msum,
                                float* __restrict__ deg,
                                int E) {
    int e = blockIdx.x * blockDim.x + threadIdx.x;
    if (e >= E) return;
    int si = src[e];
    int di = dst[e];
    float v = w[e] * rowsum[si];
    fadd_atomic(msum + di, v);
    fadd_atomic(deg + di, 1.0f);
}

// Kernel 3: info[n] = |rowsum[n] - msum[n]/max(deg[n],1)|
__global__ void nis_final_kernel(const float* __restrict__ rowsum,
                                 const float* __restrict__ msum,
                                 const float* __restrict__ deg,
                                 float* __restrict__ out, int N) {
    int n = blockIdx.x * blockDim.x + threadIdx.x;
    if (n >= N) return;
    float h1 = msum[n] / fmaxf(deg[n], 1.0f);
    out[n] = fabsf(rowsum[n] - h1);
}

extern "C" void kernel_launch(void* const* d_in, const int* in_sizes, int n_in,
                              void* d_out, int out_size, void* d_ws, size_t ws_size,
                              hipStream_t stream) {
    (void)n_in; (void)ws_size;
    const float* h   = (const float*)d_in[0];
    const float* w   = (const float*)d_in[1];
    const int*   src = (const int*)d_in[2];
    const int*   dst = (const int*)d_in[3];
    float* out = (float*)d_out;

    int N = out_size;               // output is [N]
    int E = in_sizes[1];            // weights is [E]
    int D = (N > 0) ? (int)(in_sizes[0] / (size_t)N) : 0;   // h is [N, D]

    float* ws     = (float*)d_ws;
    float* rowsum = ws;
    float* msum   = ws + (size_t)N;
    float* deg    = ws + 2 * (size_t)N;

    int ntiles  = (N + 15) / 16;
    int blocks1 = (ntiles + 7) / 8;   // 8 waves (= 8 tiles) per 256-thread block
    // blocks1*256 >= 2N, so the zeroing pass inside kernel 1 covers all N.
    nis_rowsum_kernel<<<blocks1, 256, 0, stream>>>(h, rowsum, msum, deg, N, D, ntiles);
    nis_edge_kernel<<<(E + 255) / 256, 256, 0, stream>>>(src, dst, w, rowsum, msum, deg, E);
    nis_final_kernel<<<(N + 255) / 256, 256, 0, stream>>>(rowsum, msum, deg, out, N);
}